// Transformer_16535624090174
// MI455X (gfx1250) — compile-verified
//
#include <hip/hip_runtime.h>
#include <math.h>

// Problem constants (from reference setup_inputs)
#define BB_ 8
#define SS_ 2048
#define DDim 1024
#define OOut 2
#define NCH 16
#define CSZ (SS_ / NCH)        // 128
#define INV_SCALE 0.03125f     // 1/sqrt(1024)
#define LN_EPS 1e-5f

typedef float v2f __attribute__((ext_vector_type(2)));
typedef float v8f __attribute__((ext_vector_type(8)));

__device__ __forceinline__ float fm_gelu1(float x) {
  // exact gelu(x) + 1
  return 0.5f * x * (1.f + erff(x * 0.7071067811865475f)) + 1.f;
}

// reduce modes
#define R_MUL  0   // acc += q*k                       (raw gds dot)
#define R_FMQK 1   // qf = fm(q); acc += fm(q)*fm(k)   (attn, per-s K)
#define R_SUMQ 2   // acc += q                         (cumsum phase1 / plain sum)
#define R_FMQ  3   // qf = fm(q); acc += fm(q)         (attn, broadcast K)

// ---------------- embedding gather (coalesced float4 + prefetch) ------------
__global__ void k_gather(const int* __restrict__ idx, const float* __restrict__ emb,
                         float* __restrict__ out) {
  int row = blockIdx.x;                       // 0 .. B*S-1
  int tok = idx[row];
  if (threadIdx.x == 0 && row + 512 < BB_ * SS_) {
    int tnext = idx[row + 512];
    __builtin_prefetch(emb + (size_t)tnext * DDim, 0, 1);   // global_prefetch_b8
  }
  const float4* src = (const float4*)(emb + (size_t)tok * DDim);
  float4* dst = (float4*)(out + (size_t)row * DDim);
  dst[threadIdx.x] = src[threadIdx.x];        // 256 thr * float4 = 1024
}

// ------------- chunked S-direction reduce (coalesced across d) --------------
// grid = (D/256, NCH, B); each thread owns one (b,d) within chunk c.
__global__ void k_chunk_reduce(const float* __restrict__ q, const float* __restrict__ k,
                               float* __restrict__ partial, float* __restrict__ qf,
                               int mode) {
  int d = blockIdx.x * blockDim.x + threadIdx.x;
  int c = blockIdx.y, b = blockIdx.z;
  size_t base = ((size_t)b * SS_ + (size_t)c * CSZ) * DDim + d;
  float acc = 0.f;
  for (int s = 0; s < CSZ; ++s) {
    size_t o = base + (size_t)s * DDim;
    float qv = q[o];
    float r;
    if (mode == R_MUL) {
      r = qv * k[o];
    } else if (mode == R_FMQK) {
      float g = fm_gelu1(qv); qf[o] = g; r = g * fm_gelu1(k[o]);
    } else if (mode == R_SUMQ) {
      r = qv;
    } else { // R_FMQ
      float g = fm_gelu1(qv); qf[o] = g; r = g;
    }
    acc += r;
  }
  partial[((size_t)b * NCH + c) * DDim + d] = acc;
}

// -------- cumsum phase2: add chunk-prefix offset and do sequential scan -----
__global__ void k_cumsum_apply(const float* __restrict__ x, const float* __restrict__ partial,
                               float* __restrict__ out) {
  int d = blockIdx.x * blockDim.x + threadIdx.x;
  int c = blockIdx.y, b = blockIdx.z;
  float run = 0.f;
  for (int cc = 0; cc < c; ++cc) run += partial[((size_t)b * NCH + cc) * DDim + d];
  size_t base = ((size_t)b * SS_ + (size_t)c * CSZ) * DDim + d;
  for (int s = 0; s < CSZ; ++s) {
    size_t o = base + (size_t)s * DDim;
    run += x[o];
    out[o] = run;
  }
}

// -------- finish reduce: coef[b,d] = (opt fm(k2)) * sum_c partial * scale ----
__global__ void k_finish(const float* __restrict__ partial, const float* __restrict__ k2,
                         float* __restrict__ coef, float mulscale) {
  int i = blockIdx.x * blockDim.x + threadIdx.x;  // B*D
  int b = i / DDim, d = i % DDim;
  float ssum = 0.f;
  for (int c = 0; c < NCH; ++c) ssum += partial[((size_t)b * NCH + c) * DDim + d];
  float kk = k2 ? fm_gelu1(k2[i]) : 1.f;
  coef[i] = ssum * kk * mulscale;
}

// -------- online softmax stats per chunk: running (max, scaled-sum) ---------
__global__ void k_stats_chunk(const float* __restrict__ qf, const float* __restrict__ coef,
                              float* __restrict__ mpart, float* __restrict__ lpart) {
  int d = blockIdx.x * blockDim.x + threadIdx.x;
  int c = blockIdx.y, b = blockIdx.z;
  float cf = coef[(size_t)b * DDim + d];
  size_t base = ((size_t)b * SS_ + (size_t)c * CSZ) * DDim + d;
  float m = -INFINITY, l = 0.f;
  for (int s = 0; s < CSZ; ++s) {
    float t = cf * qf[base + (size_t)s * DDim];
    float nm = fmaxf(m, t);
    l = l * expf(m - nm) + expf(t - nm);
    m = nm;
  }
  size_t pi = ((size_t)b * NCH + c) * DDim + d;
  mpart[pi] = m;
  lpart[pi] = l;
}

__global__ void k_stats_combine(const float* __restrict__ mpart, const float* __restrict__ lpart,
                                float* __restrict__ mo, float* __restrict__ lo) {
  int i = blockIdx.x * blockDim.x + threadIdx.x;  // B*D
  int b = i / DDim, d = i % DDim;
  float m = -INFINITY;
  for (int c = 0; c < NCH; ++c)
    m = fmaxf(m, mpart[((size_t)b * NCH + c) * DDim + d]);
  float l = 0.f;
  for (int c = 0; c < NCH; ++c) {
    size_t pi = ((size_t)b * NCH + c) * DDim + d;
    l += lpart[pi] * expf(mpart[pi] - m);
  }
  mo[i] = m;
  lo[i] = l;
}

// -------- out[b,s,d] = coef[b,d] * q[b,s,d]  (rank-1 gds apply) -------------
__global__ void k_scale_rows(const float* __restrict__ q, const float* __restrict__ coef,
                             float* __restrict__ out) {
  int row = blockIdx.x, b = row / SS_, tid = threadIdx.x;
  float4 qv = ((const float4*)(q + (size_t)row * DDim))[tid];
  float4 cv = ((const float4*)(coef + (size_t)b * DDim))[tid];
  float4 r;
  r.x = qv.x * cv.x; r.y = qv.y * cv.y; r.z = qv.z * cv.z; r.w = qv.w * cv.w;
  ((float4*)(out + (size_t)row * DDim))[tid] = r;
}

// -------- fused: softmax(coef*qf) * rep + residual, then LayerNorm over D ---
// One block per (b,s) row, in-place on `io`. The three bulk 4KB row streams
// are moved with CDNA5 GLOBAL_LOAD_ASYNC_TO_LDS_B128 (16B/lane, ASYNCcnt),
// overlapped with the small L2-resident per-(b,d) vector loads, then a single
// s_wait_asynccnt 0 + barrier before compute.
__global__ void k_apply_ln(const float* __restrict__ qf, const float* __restrict__ coef,
                           const float* __restrict__ mb, const float* __restrict__ lb,
                           const float* __restrict__ rep, int rep_bcast,
                           const float* __restrict__ g, const float* __restrict__ beta,
                           float* __restrict__ io) {
  __shared__ __align__(16) float s_qf[DDim];
  __shared__ __align__(16) float s_rep[DDim];
  __shared__ __align__(16) float s_rs[DDim];
  __shared__ float s_sum[256];
  __shared__ float s_sq[256];
  int row = blockIdx.x, b = row / SS_, tid = threadIdx.x;

  {
    // per-lane LDS byte addresses (low 32 bits of the shared-aperture address)
    unsigned lqf  = (unsigned)(size_t)(&s_qf[0])  + (unsigned)tid * 16u;
    unsigned lrep = (unsigned)(size_t)(&s_rep[0]) + (unsigned)tid * 16u;
    unsigned lrs  = (unsigned)(size_t)(&s_rs[0])  + (unsigned)tid * 16u;
    const float* repr = rep + (rep_bcast ? (size_t)b : (size_t)row) * DDim;
    unsigned long long gqf  = (unsigned long long)(size_t)(qf + (size_t)row * DDim) + (unsigned long long)tid * 16u;
    unsigned long long grep = (unsigned long long)(size_t)repr + (unsigned long long)tid * 16u;
    unsigned long long grs  = (unsigned long long)(size_t)(io + (size_t)row * DDim) + (unsigned long long)tid * 16u;
    asm volatile("global_load_async_to_lds_b128 %0, %1, off"
                 :: "v"(lqf), "v"(gqf) : "memory");
    asm volatile("global_load_async_to_lds_b128 %0, %1, off"
                 :: "v"(lrep), "v"(grep) : "memory");
    asm volatile("global_load_async_to_lds_b128 %0, %1, off"
                 :: "v"(lrs), "v"(grs) : "memory");
  }

  // overlap with async: small (B,D) vectors, L2-resident
  float4 cf = ((const float4*)(coef + (size_t)b * DDim))[tid];
  float4 mv = ((const float4*)(mb  + (size_t)b * DDim))[tid];
  float4 lv = ((const float4*)(lb  + (size_t)b * DDim))[tid];

  asm volatile("s_wait_asynccnt 0x0" ::: "memory");
  __syncthreads();

  float4 q  = ((const float4*)s_qf)[tid];
  float4 rv = ((const float4*)s_rep)[tid];
  float4 rs = ((const float4*)s_rs)[tid];

  float y0 = expf(cf.x * q.x - mv.x) / lv.x * rv.x + rs.x;
  float y1 = expf(cf.y * q.y - mv.y) / lv.y * rv.y + rs.y;
  float y2 = expf(cf.z * q.z - mv.z) / lv.z * rv.z + rs.z;
  float y3 = expf(cf.w * q.w - mv.w) / lv.w * rv.w + rs.w;

  s_sum[tid] = y0 + y1 + y2 + y3;
  s_sq[tid]  = y0 * y0 + y1 * y1 + y2 * y2 + y3 * y3;
  __syncthreads();
  for (int off = 128; off > 0; off >>= 1) {
    if (tid < off) { s_sum[tid] += s_sum[tid + off]; s_sq[tid] += s_sq[tid + off]; }
    __syncthreads();
  }
  float mean = s_sum[0] * (1.f / DDim);
  float var  = fmaxf(s_sq[0] * (1.f / DDim) - mean * mean, 0.f);
  float inv  = rsqrtf(var + LN_EPS);

  float4 gg = ((const float4*)g)[tid];
  float4 be = ((const float4*)beta)[tid];
  float4 o;
  o.x = (y0 - mean) * inv * gg.x + be.x;
  o.y = (y1 - mean) * inv * gg.y + be.y;
  o.z = (y2 - mean) * inv * gg.z + be.z;
  o.w = (y3 - mean) * inv * gg.w + be.w;
  ((float4*)(io + (size_t)row * DDim))[tid] = o;
}

// -------- grab ctx[:, S-1, :] into a (B,D) buffer ---------------------------
__global__ void k_last_row(const float* __restrict__ ctx, float* __restrict__ out) {
  int i = blockIdx.x * blockDim.x + threadIdx.x;  // B*D
  int b = i / DDim, d = i % DDim;
  out[i] = ctx[((size_t)b * SS_ + (SS_ - 1)) * DDim + d];
}

// -------- classifier: (8x1024)@(1024x2)+bc via V_WMMA_F32_16X16X4_F32 -------
// Single wave. M=16 (8 valid), N=16 (2 valid), K loop of 4.
// A 16x4 layout: lanes 0-15 hold K=0,1 (v0,v1) for M=lane; lanes 16-31 hold K=2,3.
// B 4x16 layout: v0 = rows {K=0 | K=2}, v1 = rows {K=1 | K=3}, N = lane&15.
// D layout: VGPR r = D[M=r][N=lane] for lanes 0-15.
__global__ void k_classifier(const float* __restrict__ pooled, const float* __restrict__ Wc,
                             const float* __restrict__ bc, float* __restrict__ out) {
#if __has_builtin(__builtin_amdgcn_wmma_f32_16x16x4_f32)
  int lane = threadIdx.x;
  int half = lane >> 4;      // 0: K pair {0,1}; 1: K pair {2,3}
  int mn = lane & 15;
  v8f c = {0.f, 0.f, 0.f, 0.f, 0.f, 0.f, 0.f, 0.f};
  for (int k0 = 0; k0 < DDim; k0 += 4) {
    int ka = k0 + half * 2;
    v2f a = {0.f, 0.f};
    v2f bmat = {0.f, 0.f};
    if (mn < BB_) {
      a.x = pooled[mn * DDim + ka];
      a.y = pooled[mn * DDim + ka + 1];
    }
    if (mn < OOut) {
      bmat.x = Wc[ka * OOut + mn];
      bmat.y = Wc[(ka + 1) * OOut + mn];
    }
    c = __builtin_amdgcn_wmma_f32_16x16x4_f32(false, a, false, bmat,
                                              (short)0, c, false, false);
  }
  if (lane < OOut) {
#pragma unroll
    for (int r = 0; r < BB_; ++r) out[r * OOut + lane] = c[r] + bc[lane];
  }
#else
  // scalar fallback
  int lane = threadIdx.x;
  if (lane < BB_ * OOut) {
    int m = lane / OOut, n = lane % OOut;
    float acc = bc[n];
    for (int k = 0; k < DDim; ++k) acc += pooled[m * DDim + k] * Wc[k * OOut + n];
    out[lane] = acc;
  }
#endif
}

extern "C" void kernel_launch(void* const* d_in, const int* in_sizes, int n_in,
                              void* d_out, int out_size, void* d_ws, size_t ws_size,
                              hipStream_t stream) {
  (void)in_sizes; (void)n_in; (void)out_size; (void)ws_size;
  const int*   src    = (const int*)d_in[0];
  const int*   trg    = (const int*)d_in[1];
  const float* emb    = (const float*)d_in[2];
  const float* enc_g  = (const float*)d_in[3];
  const float* enc_b  = (const float*)d_in[4];
  const float* dec_ng = (const float*)d_in[5];
  const float* dec_nb = (const float*)d_in[6];
  const float* dec_tg = (const float*)d_in[7];
  const float* dec_tb = (const float*)d_in[8];
  const float* Wc     = (const float*)d_in[9];
  const float* bc     = (const float*)d_in[10];
  float* outp = (float*)d_out;

  // workspace carve-up (floats)
  float* w = (float*)d_ws;
  const size_t BSD = (size_t)BB_ * SS_ * DDim;
  float* t0 = w; w += BSD;                 // x / xe
  float* t1 = w; w += BSD;                 // ctx
  float* t2 = w; w += BSD;                 // out / q / x  (running state)
  float* qf = w; w += BSD;                 // fm(q) cache per attn stage
  float* t4 = w; w += BSD;                 // rep (layer 1)
  float* partial = w; w += (size_t)BB_ * NCH * DDim;
  float* mpart   = w; w += (size_t)BB_ * NCH * DDim;
  float* lpart   = w; w += (size_t)BB_ * NCH * DDim;
  float* coef   = w; w += (size_t)BB_ * DDim;
  float* mbuf   = w; w += (size_t)BB_ * DDim;
  float* lbuf   = w; w += (size_t)BB_ * DDim;
  float* k2raw  = w; w += (size_t)BB_ * DDim;
  float* v2     = w; w += (size_t)BB_ * DDim;
  float* pooled = w; w += (size_t)BB_ * DDim;

  const dim3 gc(DDim / 256, NCH, BB_);     // chunked S-direction kernels
  const int rows = BB_ * SS_;              // per-row kernels
  const int bd_blocks = (BB_ * DDim) / 256;

  auto cumsum = [&](const float* x, float* o) {
    k_chunk_reduce<<<gc, 256, 0, stream>>>(x, nullptr, partial, nullptr, R_SUMQ);
    k_cumsum_apply<<<gc, 256, 0, stream>>>(x, partial, o);
  };
  auto gds = [&](const float* q, const float* k, float* o) {
    k_chunk_reduce<<<gc, 256, 0, stream>>>(q, k, partial, nullptr, R_MUL);
    k_finish<<<bd_blocks, 256, 0, stream>>>(partial, nullptr, coef, 1.0f);
    k_scale_rows<<<rows, 256, 0, stream>>>(q, coef, o);
  };
  // attn(q, k[, bcast], v=rep[, bcast]) + residual + LN, in-place on `io`(=q)
  auto attn_stage = [&](float* io, const float* kmat, const float* k2,
                        const float* rep, int rep_bcast,
                        const float* g, const float* bvec) {
    k_chunk_reduce<<<gc, 256, 0, stream>>>(io, kmat, partial, qf,
                                           kmat ? R_FMQK : R_FMQ);
    k_finish<<<bd_blocks, 256, 0, stream>>>(partial, k2, coef, INV_SCALE);
    k_stats_chunk<<<gc, 256, 0, stream>>>(qf, coef, mpart, lpart);
    k_stats_combine<<<bd_blocks, 256, 0, stream>>>(mpart, lpart, mbuf, lbuf);
    k_apply_ln<<<rows, 256, 0, stream>>>(qf, coef, mbuf, lbuf, rep, rep_bcast,
                                         g, bvec, io);
  };

  // ======================= encoder =======================
  k_gather<<<rows, 256, 0, stream>>>(src, emb, t0);
  cumsum(t0, t1);                              // ctx
  gds(t0, t1, t2);                             // out = gds(x, ctx); rep == out
  // layer 0
  attn_stage(t2, t1, nullptr, t2, 0, enc_g, enc_b);
  // inter-layer: ctx = cumsum(out); rep = gds(out, ctx)
  cumsum(t2, t1);
  gds(t2, t1, t4);
  // layer 1
  attn_stage(t2, t1, nullptr, t4, 0, enc_g + DDim, enc_b + DDim);
  // enc_ctx last row -> k2raw ; v2 = sum_s enc_out
  k_last_row<<<bd_blocks, 256, 0, stream>>>(t1, k2raw);
  k_chunk_reduce<<<gc, 256, 0, stream>>>(t2, nullptr, partial, nullptr, R_SUMQ);
  k_finish<<<bd_blocks, 256, 0, stream>>>(partial, nullptr, v2, 1.0f);

  // ======================= decoder =======================
  k_gather<<<rows, 256, 0, stream>>>(trg, emb, t0);
  cumsum(t0, t1);
  gds(t0, t1, t2);                             // x = gds(xe, ctx); rep == x
  // layer 0
  attn_stage(t2, t1, nullptr, t2, 0, dec_ng, dec_nb);            // self
  attn_stage(t2, nullptr, k2raw, v2, 1, dec_tg, dec_tb);         // cross (bcast)
  cumsum(t2, t1);
  gds(t2, t1, t4);
  // layer 1
  attn_stage(t2, t1, nullptr, t4, 0, dec_ng + DDim, dec_nb + DDim);
  attn_stage(t2, nullptr, k2raw, v2, 1, dec_tg + DDim, dec_tb + DDim);

  // pooled = mean over S, then WMMA classifier
  k_chunk_reduce<<<gc, 256, 0, stream>>>(t2, nullptr, partial, nullptr, R_SUMQ);
  k_finish<<<bd_blocks, 256, 0, stream>>>(partial, nullptr, pooled, 1.0f / SS_);
  k_classifier<<<1, 32, 0, stream>>>(pooled, Wc, bc, outp);
}